// ConvCaps_34995393528409
// MI455X (gfx1250) — compile-verified
//
#include <hip/hip_runtime.h>

typedef __attribute__((ext_vector_type(2))) float v2f;
typedef __attribute__((ext_vector_type(8))) float v8f;

// Fixed problem shape (from reference setup_inputs):
//   poses  : (B=32, H=20, W=20, I=32, M=4, N=4)  f32
//   kernel : (KX=3, KY=3, I=32, O=32, M=4, N=4)  f32
//   out    : (B=32, OH=18, OW=18, KX=3, KY=3, O=32, M=4, N=4) f32
// einsum contracts over i only -> 144 independent [10368 x 32] x [32 x 32] GEMMs.
#define B_N   32
#define H_N   20
#define W_N   20
#define I_N   32
#define O_N   32
#define MN_N  16          // M*N = 4*4 pose elements
#define OH_N  18
#define OW_N  18
#define ROWS  (B_N * OH_N * OW_N)          // 10368 GEMM rows
#define POSE_ROW_STRIDE (I_N * MN_N)       // 512 floats per (b,hh,ww)
#define OUT_ROW_STRIDE  (3 * 3 * O_N * MN_N) // 4608 floats per (b,h,w)
#define LDS_KSTRIDE 48                      // padded row stride: half-wave K rows
                                            // differ by 2*48=96 -> disjoint bank groups

__global__ __launch_bounds__(256) void convcaps_wmma_f32(
    const float* __restrict__ poses,
    const float* __restrict__ kern,
    float* __restrict__ out)
{
    const int xy = blockIdx.y;          // 0..8  -> (x,y) kernel offset
    const int mn = blockIdx.z;          // 0..15 -> (m,n) pose element
    const int kx = xy / 3;
    const int ky = xy % 3;

    // Stage kernel[x,y,:,:,m,n] as ldsK[i*48 + o]  (32x32 floats, padded rows)
    __shared__ float ldsK[I_N * LDS_KSTRIDE];
    {
        const float* kbase = kern + (size_t)xy * I_N * O_N * MN_N + mn;
        const int tid = threadIdx.x;
        #pragma unroll
        for (int j = 0; j < 4; ++j) {
            int e = tid * 4 + j;               // 0..1023 flat (i,o)
            int i = e >> 5;
            int o = e & 31;
            ldsK[i * LDS_KSTRIDE + o] = kbase[(size_t)e * MN_N];
        }
    }
    __syncthreads();

    const int wave  = threadIdx.x >> 5;     // 0..7
    const int lane  = threadIdx.x & 31;
    const int lhalf = lane >> 4;            // 0: K pair (0,1); 1: K pair (2,3)
    const int l16   = lane & 15;

    const int tile = blockIdx.x * 8 + wave; // 0..647 (16 rows each)
    const int r0   = tile * 16;

    // This lane's A-matrix row (M = l16): flat output row r -> (b,h,w)
    const int r  = r0 + l16;
    const int w  = r % OW_N;
    const int hw = r / OW_N;
    const int h  = hw % OH_N;
    const int b  = hw / OH_N;

    const float* prow = poses
        + (((size_t)b * H_N + (h + kx)) * W_N + (w + ky)) * POSE_ROW_STRIDE
        + mn;
    __builtin_prefetch(prow, 0, 1);         // global_prefetch_b8 on gfx1250

    // A fragments for all 8 K-steps (ISA 16x4 f32 layout:
    // lanes 0-15 hold K=4k,4k+1; lanes 16-31 hold K=4k+2,4k+3; row M = lane&15).
    v2f a[8];
    #pragma unroll
    for (int kk = 0; kk < 8; ++kk) {
        int i0 = kk * 4 + lhalf * 2;
        a[kk].x = prow[(size_t)i0 * MN_N];
        a[kk].y = prow[(size_t)(i0 + 1) * MN_N];
    }

    // Two 16x16 accumulators: o in [0,16) and [16,32)
    v8f c0 = {};
    v8f c1 = {};
    #pragma unroll
    for (int kk = 0; kk < 8; ++kk) {
        int k0 = kk * 4 + lhalf * 2;
        v2f b0, b1;
        b0.x = ldsK[k0 * LDS_KSTRIDE + l16];
        b0.y = ldsK[(k0 + 1) * LDS_KSTRIDE + l16];
        b1.x = ldsK[k0 * LDS_KSTRIDE + 16 + l16];
        b1.y = ldsK[(k0 + 1) * LDS_KSTRIDE + 16 + l16];
        c0 = __builtin_amdgcn_wmma_f32_16x16x4_f32(
                false, a[kk], false, b0, (short)0, c0, false, false);
        c1 = __builtin_amdgcn_wmma_f32_16x16x4_f32(
                false, a[kk], false, b1, (short)0, c1, false, false);
    }

    // D layout: lane -> N = l16 (+16 for c1); VGPR v -> M = v + 8*lhalf.
    // out flat = r*4608 + xy*512 + o*16 + mn  (b,h,w contiguous => linear in r)
    #pragma unroll
    for (int v = 0; v < 8; ++v) {
        int    rm    = r0 + v + 8 * lhalf;
        size_t obase = (size_t)rm * OUT_ROW_STRIDE + (size_t)xy * (O_N * MN_N) + mn;
        out[obase + (size_t)l16 * MN_N]        = c0[v];
        out[obase + (size_t)(l16 + 16) * MN_N] = c1[v];
    }
}

extern "C" void kernel_launch(void* const* d_in, const int* in_sizes, int n_in,
                              void* d_out, int out_size, void* d_ws, size_t ws_size,
                              hipStream_t stream)
{
    const float* poses = (const float*)d_in[0];
    // d_in[1] (activations) is unused by the reference einsum.
    const float* kern  = (const float*)d_in[2];
    float* out = (float*)d_out;

    // 81 row-groups (128 GEMM rows each) x 9 kernel offsets x 16 pose elements
    dim3 grid(81, 9, 16);
    convcaps_wmma_f32<<<grid, 256, 0, stream>>>(poses, kern, out);
    (void)in_sizes; (void)n_in; (void)out_size; (void)d_ws; (void)ws_size;
}